// GATReasoner_64518998720686
// MI455X (gfx1250) — compile-verified
//
#include <hip/hip_runtime.h>
#include <math.h>

typedef __attribute__((ext_vector_type(2))) float v2f;
typedef __attribute__((ext_vector_type(8))) float v8f;

#define NEG_SLOPE 0.2f
#define RECON_W   0.1f
#define BN_EPS    1e-5f

// ---- monotone uint encoding of float for integer atomicMax ----
__device__ __forceinline__ unsigned f2ord(float f) {
    unsigned u = __float_as_uint(f);
    return (u & 0x80000000u) ? ~u : (u | 0x80000000u);
}
__device__ __forceinline__ float ord2f(unsigned u) {
    unsigned b = (u & 0x80000000u) ? (u ^ 0x80000000u) : ~u;
    return __uint_as_float(b);
}
__device__ __forceinline__ float leaky(float x) { return x > 0.0f ? x : x * NEG_SLOPE; }
__device__ __forceinline__ void atomAddF(float* p, float v) {
    __hip_atomic_fetch_add(p, v, __ATOMIC_RELAXED, __HIP_MEMORY_SCOPE_AGENT);
}

// ============================================================
// WMMA fp32 GEMM: C[M,NCOL] = A[M,K] @ B[K,NCOL], row-major.
// K, NCOL compile-time -> base pointer + immediate-offset loads,
// fully unrolled K loop of V_WMMA_F32_16X16X4_F32.
// Block = 32 * COL_TILES * ROW_TILES threads; wave = one 16x16 tile.
// ============================================================
template<int K, int NCOL, int COL_TILES, int ROW_TILES>
__global__ void gemm_wmma_f32(const float* __restrict__ A, const float* __restrict__ B,
                              float* __restrict__ C, int M) {
    const int wid  = threadIdx.x >> 5;
    const int lane = threadIdx.x & 31;
    const int rowsub = wid / COL_TILES;
    const int colt   = wid % COL_TILES;
    const int row0 = (blockIdx.x * ROW_TILES + rowsub) * 16;
    const int col0 = colt * 16;
    if (row0 >= M) return;                 // wave-uniform exit, EXEC stays full

    const int m   = lane & 15;
    const int sel = lane >> 4;             // 0 -> K+0/1 ; 1 -> K+2/3 (ISA 16x4 layout)
    const int row = row0 + m;
    const int rowc = row < M ? row : M - 1;  // clamp: EXEC must stay all-ones for WMMA

    const float* __restrict__ Ap = A + (size_t)rowc * K + (sel << 1);
    const float* __restrict__ Bp = B + (size_t)(sel << 1) * NCOL + col0 + m;

    v8f acc = {0.f, 0.f, 0.f, 0.f, 0.f, 0.f, 0.f, 0.f};
#pragma unroll
    for (int k0 = 0; k0 < K; k0 += 4) {
        v2f a, b;
        a.x = Ap[k0];                      // consecutive -> global_load_b64 offset:imm
        a.y = Ap[k0 + 1];
        b.x = Bp[k0 * NCOL];               // immediate offsets, single base pointer
        b.y = Bp[k0 * NCOL + NCOL];
        acc = __builtin_amdgcn_wmma_f32_16x16x4_f32(false, a, false, b,
                                                    (short)0, acc, false, false);
    }
    const int col = col0 + m;
#pragma unroll
    for (int i = 0; i < 8; ++i) {          // VGPR i -> rows i (lanes 0-15), i+8 (16-31)
        const int r = row0 + i + (sel ? 8 : 0);
        if (r < M) C[(size_t)r * NCOL + col] = acc[i];
    }
}

// ============================================================
// Layer-1 attention coefficients + self-loop max seed
// thread t = n*4 + head
// ============================================================
__global__ void alpha1_kernel(const float* __restrict__ h1,
                              const float* __restrict__ a_src, const float* __restrict__ a_dst,
                              float* __restrict__ asrc, float* __restrict__ adst,
                              unsigned* __restrict__ m1, int N) {
    int t = blockIdx.x * blockDim.x + threadIdx.x;
    if (t >= N * 4) return;
    int h = t & 3, n = t >> 2;
    const float* hp = h1 + (size_t)n * 128 + h * 32;
    const float* as = a_src + h * 32;
    const float* ad = a_dst + h * 32;
    float s = 0.f, d = 0.f;
#pragma unroll 8
    for (int c = 0; c < 32; ++c) { float v = hp[c]; s += v * as[c]; d += v * ad[c]; }
    asrc[t] = s; adst[t] = d;
    m1[t] = f2ord(leaky(s + d));           // self-loop seeds the segment max
}

__global__ void edgemax1_kernel(const int* __restrict__ ei,
                                const float* __restrict__ asrc, const float* __restrict__ adst,
                                unsigned* __restrict__ m1, int E) {
    int e = blockIdx.x * blockDim.x + threadIdx.x;
    if (e >= E) return;
    int s = ei[e], d = ei[E + e];
#pragma unroll
    for (int h = 0; h < 4; ++h)
        atomicMax(&m1[d * 4 + h], f2ord(leaky(asrc[s * 4 + h] + adst[d * 4 + h])));
}

// out1 <- w_self * h1 ; s1 <- w_self     (thread per (n, channel))
__global__ void initacc1_kernel(const float* __restrict__ h1,
                                const float* __restrict__ asrc, const float* __restrict__ adst,
                                const unsigned* __restrict__ m1,
                                float* __restrict__ s1, float* __restrict__ out1, int N) {
    long long t = (long long)blockIdx.x * blockDim.x + threadIdx.x;
    if (t >= (long long)N * 128) return;
    int c = (int)(t & 127); int n = (int)(t >> 7);
    int h = c >> 5;
    float e = leaky(asrc[n * 4 + h] + adst[n * 4 + h]);
    float w = expf(e - ord2f(m1[n * 4 + h]));
    out1[t] = w * h1[t];
    if ((c & 31) == 0) s1[n * 4 + h] = w;
}

// one wave per edge; lane covers 4 channels (same head), L2-resident f32 atomics
__global__ void edgeagg1_kernel(const int* __restrict__ ei, const float* __restrict__ h1,
                                const float* __restrict__ asrc, const float* __restrict__ adst,
                                const unsigned* __restrict__ m1,
                                float* __restrict__ s1, float* __restrict__ out1, int E) {
    int lane = threadIdx.x & 31;
    int e = blockIdx.x * (blockDim.x >> 5) + (threadIdx.x >> 5);
    if (e >= E) return;
    int s = ei[e], d = ei[E + e];
    int h = lane >> 3;                      // channels [4*lane,4*lane+3] share head lane/8
    float x = leaky(asrc[s * 4 + h] + adst[d * 4 + h]);
    float w = expf(x - ord2f(m1[d * 4 + h]));
    const float4 hv = *(const float4*)(h1 + (size_t)s * 128 + lane * 4);
    float* op = out1 + (size_t)d * 128 + lane * 4;
    atomAddF(op + 0, w * hv.x);
    atomAddF(op + 1, w * hv.y);
    atomAddF(op + 2, w * hv.z);
    atomAddF(op + 3, w * hv.w);
    if ((lane & 7) == 0) atomAddF(&s1[d * 4 + h], w);
}

// normalize + bias + BN(eval) + ELU, in place (out1 becomes g)
__global__ void final1_kernel(const float* __restrict__ s1, const float* __restrict__ b1,
                              const float* __restrict__ gam, const float* __restrict__ bet,
                              const float* __restrict__ mean, const float* __restrict__ var,
                              float* __restrict__ out1, int N) {
    long long t = (long long)blockIdx.x * blockDim.x + threadIdx.x;
    if (t >= (long long)N * 128) return;
    int c = (int)(t & 127); int n = (int)(t >> 7);
    float v = out1[t] / s1[n * 4 + (c >> 5)] + b1[c];
    v = (v - mean[c]) * rsqrtf(var[c] + BN_EPS) * gam[c] + bet[c];
    out1[t] = v > 0.f ? v : expm1f(v);      // ELU
}

// ============================================================
// Layer 2 (1 head, C=32)
// ============================================================
__global__ void alpha2_kernel(const float* __restrict__ h2,
                              const float* __restrict__ a_src, const float* __restrict__ a_dst,
                              float* __restrict__ asrc, float* __restrict__ adst,
                              unsigned* __restrict__ m2, int N) {
    int n = blockIdx.x * blockDim.x + threadIdx.x;
    if (n >= N) return;
    const float* hp = h2 + (size_t)n * 32;
    float s = 0.f, d = 0.f;
#pragma unroll 8
    for (int c = 0; c < 32; ++c) { float v = hp[c]; s += v * a_src[c]; d += v * a_dst[c]; }
    asrc[n] = s; adst[n] = d;
    m2[n] = f2ord(leaky(s + d));
}

__global__ void edgemax2_kernel(const int* __restrict__ ei,
                                const float* __restrict__ asrc, const float* __restrict__ adst,
                                unsigned* __restrict__ m2, int E) {
    int e = blockIdx.x * blockDim.x + threadIdx.x;
    if (e >= E) return;
    int s = ei[e], d = ei[E + e];
    atomicMax(&m2[d], f2ord(leaky(asrc[s] + adst[d])));
}

__global__ void initacc2_kernel(const float* __restrict__ h2,
                                const float* __restrict__ asrc, const float* __restrict__ adst,
                                const unsigned* __restrict__ m2,
                                float* __restrict__ s2, float* __restrict__ out2, int N) {
    long long t = (long long)blockIdx.x * blockDim.x + threadIdx.x;
    if (t >= (long long)N * 32) return;
    int c = (int)(t & 31); int n = (int)(t >> 5);
    float w = expf(leaky(asrc[n] + adst[n]) - ord2f(m2[n]));
    out2[t] = w * h2[t];
    if (c == 0) s2[n] = w;
}

__global__ void edgeagg2_kernel(const int* __restrict__ ei, const float* __restrict__ h2,
                                const float* __restrict__ asrc, const float* __restrict__ adst,
                                const unsigned* __restrict__ m2,
                                float* __restrict__ s2, float* __restrict__ out2, int E) {
    int lane = threadIdx.x & 31;
    int e = blockIdx.x * (blockDim.x >> 5) + (threadIdx.x >> 5);
    if (e >= E) return;
    int s = ei[e], d = ei[E + e];
    float w = expf(leaky(asrc[s] + adst[d]) - ord2f(m2[d]));
    atomAddF(&out2[(size_t)d * 32 + lane], w * h2[(size_t)s * 32 + lane]);
    if (lane == 0) atomAddF(&s2[d], w);
}

// ============================================================
// Tail: normalize layer-2, reasoner MLP, decoder, classifier, log_softmax
// one thread per node; writes logits [N,10] then expl [N,32] into d_out
// ============================================================
__global__ void tail_kernel(const float* __restrict__ out2, const float* __restrict__ s2,
                            const float* __restrict__ b2,  const float* __restrict__ ctx,
                            const float* __restrict__ rw1, const float* __restrict__ rb1,
                            const float* __restrict__ rw2, const float* __restrict__ rb2,
                            const float* __restrict__ dw,  const float* __restrict__ db,
                            const float* __restrict__ cw,  const float* __restrict__ cb,
                            float* __restrict__ out, int N) {
    int n = blockIdx.x * blockDim.x + threadIdx.x;
    if (n >= N) return;
    float c6[6];
#pragma unroll
    for (int k = 0; k < 6; ++k) c6[k] = ctx[(size_t)n * 6 + k];
    float hid[32];
#pragma unroll
    for (int j = 0; j < 32; ++j) {
        float v = rb1[j];
#pragma unroll
        for (int k = 0; k < 6; ++k) v += c6[k] * rw1[k * 32 + j];
        hid[j] = v > 0.f ? v : 0.f;
    }
    float ex[32];
#pragma unroll
    for (int j = 0; j < 32; ++j) {
        float v = rb2[j];
        for (int k = 0; k < 32; ++k) v += hid[k] * rw2[k * 32 + j];
        ex[j] = v;
    }
    float inv_s = 1.0f / s2[n];
    float comb[32];
#pragma unroll
    for (int j = 0; j < 32; ++j) {
        float rec = db[j];
        for (int k = 0; k < 32; ++k) rec += ex[k] * dw[k * 32 + j];
        comb[j] = out2[(size_t)n * 32 + j] * inv_s + b2[j] + RECON_W * rec;
    }
    float lg[10]; float mx = -INFINITY;
#pragma unroll
    for (int c = 0; c < 10; ++c) {
        float v = cb[c];
        for (int j = 0; j < 32; ++j) v += comb[j] * cw[j * 10 + c];
        lg[c] = v; mx = v > mx ? v : mx;
    }
    float ssum = 0.f;
#pragma unroll
    for (int c = 0; c < 10; ++c) ssum += expf(lg[c] - mx);
    float lse = mx + logf(ssum);
#pragma unroll
    for (int c = 0; c < 10; ++c) out[(size_t)n * 10 + c] = lg[c] - lse;
#pragma unroll
    for (int j = 0; j < 32; ++j) out[(size_t)N * 10 + (size_t)n * 32 + j] = ex[j];
}

// ============================================================
extern "C" void kernel_launch(void* const* d_in, const int* in_sizes, int n_in,
                              void* d_out, int out_size, void* d_ws, size_t ws_size,
                              hipStream_t stream) {
    const float* x      = (const float*)d_in[0];
    const int*   ei     = (const int*)d_in[1];
    const float* ctx    = (const float*)d_in[2];
    const float* W1     = (const float*)d_in[3];
    const float* a_src1 = (const float*)d_in[4];
    const float* a_dst1 = (const float*)d_in[5];
    const float* b1     = (const float*)d_in[6];
    const float* bn_g   = (const float*)d_in[7];
    const float* bn_b   = (const float*)d_in[8];
    const float* bn_m   = (const float*)d_in[9];
    const float* bn_v   = (const float*)d_in[10];
    const float* W2     = (const float*)d_in[11];
    const float* a_src2 = (const float*)d_in[12];
    const float* a_dst2 = (const float*)d_in[13];
    const float* b2     = (const float*)d_in[14];
    const float* rw1    = (const float*)d_in[15];
    const float* rb1    = (const float*)d_in[16];
    const float* rw2    = (const float*)d_in[17];
    const float* rb2    = (const float*)d_in[18];
    const float* dw     = (const float*)d_in[19];
    const float* db     = (const float*)d_in[20];
    const float* cw     = (const float*)d_in[21];
    const float* cb     = (const float*)d_in[22];

    const int N = in_sizes[0] / 128;
    const int E = in_sizes[1] / 2;

    float* ws = (float*)d_ws;
    size_t off = 0;
    float*    h1    = ws + off; off += (size_t)N * 128;
    float*    out1  = ws + off; off += (size_t)N * 128;   // becomes g after final1
    float*    asrc1 = ws + off; off += (size_t)N * 4;
    float*    adst1 = ws + off; off += (size_t)N * 4;
    unsigned* m1    = (unsigned*)(ws + off); off += (size_t)N * 4;
    float*    s1    = ws + off; off += (size_t)N * 4;
    float*    h2    = ws + off; off += (size_t)N * 32;
    float*    out2  = ws + off; off += (size_t)N * 32;
    float*    asrc2 = ws + off; off += (size_t)N;
    float*    adst2 = ws + off; off += (size_t)N;
    unsigned* m2    = (unsigned*)(ws + off); off += (size_t)N;
    float*    s2    = ws + off; off += (size_t)N;

    const int TB = 256;
    // ---- Layer 1 ----
    gemm_wmma_f32<128, 128, 8, 1><<<(N + 15) / 16, 256, 0, stream>>>(x, W1, h1, N);
    alpha1_kernel<<<(N * 4 + TB - 1) / TB, TB, 0, stream>>>(h1, a_src1, a_dst1, asrc1, adst1, m1, N);
    edgemax1_kernel<<<(E + TB - 1) / TB, TB, 0, stream>>>(ei, asrc1, adst1, m1, E);
    initacc1_kernel<<<(unsigned)(((long long)N * 128 + TB - 1) / TB), TB, 0, stream>>>(
        h1, asrc1, adst1, m1, s1, out1, N);
    edgeagg1_kernel<<<(E + 7) / 8, TB, 0, stream>>>(ei, h1, asrc1, adst1, m1, s1, out1, E);
    final1_kernel<<<(unsigned)(((long long)N * 128 + TB - 1) / TB), TB, 0, stream>>>(
        s1, b1, bn_g, bn_b, bn_m, bn_v, out1, N);
    // ---- Layer 2 ----
    gemm_wmma_f32<128, 32, 2, 4><<<(N + 63) / 64, 256, 0, stream>>>(out1, W2, h2, N);
    alpha2_kernel<<<(N + TB - 1) / TB, TB, 0, stream>>>(h2, a_src2, a_dst2, asrc2, adst2, m2, N);
    edgemax2_kernel<<<(E + TB - 1) / TB, TB, 0, stream>>>(ei, asrc2, adst2, m2, E);
    initacc2_kernel<<<(unsigned)(((long long)N * 32 + TB - 1) / TB), TB, 0, stream>>>(
        h2, asrc2, adst2, m2, s2, out2, N);
    edgeagg2_kernel<<<(E + 7) / 8, TB, 0, stream>>>(ei, h2, asrc2, adst2, m2, s2, out2, E);
    // ---- Tail ----
    tail_kernel<<<(N + TB - 1) / TB, TB, 0, stream>>>(out2, s2, b2, ctx, rw1, rb1, rw2, rb2,
                                                      dw, db, cw, cb, (float*)d_out, N);
}